// Generator_66829691126231
// MI455X (gfx1250) — compile-verified
//
#include <hip/hip_runtime.h>
#include <math.h>

// ---------------------------------------------------------------------------
// AMM forward for MI455X (gfx1250, wave32).
//
// scores_b = theta_b^T (392x4096) . phi_b (392x4096), softmax(200*s),
// weighted row reductions against g/beta vectors -- fused flash-style
// (201MB score tensor never materialized).
//
// CDNA5 mapping:
//  * fp32 WMMA (V_WMMA_F32_16X16X4_F32): the x200 softmax scale amplifies
//    f16/bf16 input rounding into O(10) exponent errors, so the fp32 matrix
//    pipe is the correct precision choice.
//  * Tensor Data Mover (TENSOR_LOAD_TO_LDS): phi key-tiles are strided 2D
//    tiles (16 cols x 392 rows, row stride 4096 floats) -> one descriptor
//    per sub-tensor, double-buffered in LDS, tracked via TENSORcnt.
//  * Block-cooperative phi stream: 4 waves share each TDM-fetched tile and
//    each owns a 16-row query sub-tile -> 4x less L2 traffic than private
//    streams and no cross-wave softmax merge.
// ---------------------------------------------------------------------------

typedef float        v2f __attribute__((ext_vector_type(2)));
typedef float        v8f __attribute__((ext_vector_type(8)));
typedef unsigned int v4u __attribute__((ext_vector_type(4)));
typedef int          v8i __attribute__((ext_vector_type(8)));
typedef int          v4i __attribute__((ext_vector_type(4)));

#define HW    4096
#define KTOT  392      // 256 feature channels + 136 rel-pos channels
#define TPAD  402      // theta LDS row stride (even -> b64 align, bank-spread)

// Butterfly reductions within a 16-lane half (rows live per-half in WMMA C/D)
__device__ __forceinline__ float bf_max16(float v) {
#pragma unroll
    for (int m = 1; m < 16; m <<= 1) v = fmaxf(v, __shfl_xor(v, m, 32));
    return v;
}
__device__ __forceinline__ float bf_sum16(float v) {
#pragma unroll
    for (int m = 1; m < 16; m <<= 1) v += __shfl_xor(v, m, 32);
    return v;
}

// ---------------------------------------------------------------------------
// TDM: load a [rows x 16] fp32 tile (row stride 4096 floats) into LDS,
// packed contiguously (16 floats per row).  D# packing per CDNA5 ISA §8.
// Toolchain here is clang-23 / therock-10.0: 6-arg builtin form.
// ---------------------------------------------------------------------------
__device__ __forceinline__ void tdm_load_tile(unsigned int lds_off,
                                              const float* gsrc,
                                              unsigned int rows) {
    const unsigned long long ga = (unsigned long long)(uintptr_t)gsrc;
    v4u g0;
    g0.x = 1u;                                            // count=1, user mode
    g0.y = lds_off;                                       // lds_addr (bytes)
    g0.z = (unsigned int)ga;                              // global_addr[31:0]
    g0.w = (unsigned int)((ga >> 32) & 0x01FFFFFFu)       // global_addr[56:32]
           | 0x80000000u;                                 // type=2 ("image")
    v8i g1;
    g1[0] = 0x00020000;                 // wg_mask=0, data_size=2 (4 bytes)
    g1[1] = (int)(4096u << 16);         // tensor_dim0[15:0]=4096 in [63:48]
    g1[2] = (int)(rows << 16);          // tensor_dim0 hi=0; tensor_dim1 lo=rows
    g1[3] = (int)(16u << 16);           // tensor_dim1 hi=0; tile_dim0=16
    g1[4] = (int)rows;                  // tile_dim1=rows; tile_dim2=0
    g1[5] = 4096;                       // tensor_dim0_stride[31:0] (elements)
    g1[6] = 0;                          // stride hi, tensor_dim1_stride lo
    g1[7] = 0;
    const v4i z4 = {0, 0, 0, 0};
    const v8i z8 = {0, 0, 0, 0, 0, 0, 0, 0};
    __builtin_amdgcn_tensor_load_to_lds(g0, g1, z4, z4, z8, 0); // cpol=0
}

// ---------------------------------------------------------------------------
// Kernel 1: old_gamma/old_beta (1x1 conv = 256-ch dot), g/b vectors, zero acc.
// ---------------------------------------------------------------------------
__global__ __launch_bounds__(256) void amm_prep_kernel(
    const float* __restrict__ fm_reference,   // (256,4096)
    const float* __restrict__ mask_ref,       // (3,4096)
    const float* __restrict__ w_lambda,       // (256,)
    const float* __restrict__ w_beta,         // (256,)
    float* __restrict__ gvec,                 // (3,4096)  old_gamma*mask_ref
    float* __restrict__ bvec,                 // (3,4096)  old_beta *mask_ref
    float* __restrict__ gamma,                // (4096,)   zeroed accumulator
    float* __restrict__ beta)                 // (4096,)
{
    __shared__ float swl[256], swb[256];
    const int tid = threadIdx.x;
    swl[tid] = w_lambda[tid];
    swb[tid] = w_beta[tid];
    __syncthreads();

    const int s = blockIdx.x * 256 + tid;
    float og = 0.f, ob = 0.f;
#pragma unroll 4
    for (int c = 0; c < 256; ++c) {
        const float v = fm_reference[c * HW + s];
        og = fmaf(swl[c], v, og);
        ob = fmaf(swb[c], v, ob);
    }
    gamma[s] = 0.f;
    beta[s]  = 0.f;
#pragma unroll
    for (int b = 0; b < 3; ++b) {
        const float mr = mask_ref[b * HW + s];
        gvec[b * HW + s] = og * mr;
        bvec[b * HW + s] = ob * mr;
    }
}

// ---------------------------------------------------------------------------
// Kernel 2: fused scores-GEMM (fp32 WMMA) + streaming softmax + g/b reduce.
// Grid: 192 blocks = (b in 0..2) x (64 query tiles of 64 rows).
// Block: 128 threads = 4 waves; wave w owns query rows [t0+16w, t0+16w+16).
// All waves share a double-buffered TDM-fed phi tile per key step.
// ---------------------------------------------------------------------------
__global__ __launch_bounds__(128) void amm_attn_kernel(
    const float* __restrict__ fm_source,      // (256,4096)
    const float* __restrict__ fm_reference,   // (256,4096)
    const float* __restrict__ mask_source,    // (3,4096)
    const float* __restrict__ mask_ref,       // (3,4096)
    const float* __restrict__ rel_pos_source, // (3,136,4096)
    const float* __restrict__ rel_pos_ref,    // (3,136,4096)
    const float* __restrict__ gvec,           // (3,4096)
    const float* __restrict__ bvec,           // (3,4096)
    float* __restrict__ gamma,                // (4096,) accumulated over b
    float* __restrict__ beta)
{
    __shared__ float theta[64 * TPAD];        // theta^T: [t][k], 64 x 392
    __shared__ float phib[2][KTOT * 16];      // phi tiles: [k][s], raw values

    const int tt   = blockIdx.x & 63;         // query 64-row tile index
    const int b    = blockIdx.x >> 6;         // region batch
    const int t0   = tt * 64;
    const int tid  = threadIdx.x;
    const int lane = tid & 31;
    const int wave = tid >> 5;
    const int half = lane >> 4;               // 0: K={0,1}, 1: K={2,3}
    const int l16  = lane & 15;               // M (A) or N (B/C/D) index

    const float* rpr = rel_pos_ref + b * 136 * HW;

    // Stage theta tile (K along fast LDS axis so A is one ds_load_b64).
    for (int n = tid; n < KTOT * 64; n += 128) {
        const int k = n >> 6, t = n & 63;
        float v;
        if (k < 256)
            v = fm_source[k * HW + t0 + t] * (0.01f * mask_source[b * HW + t0 + t]);
        else
            v = rel_pos_source[(b * 136 + (k - 256)) * HW + t0 + t];
        theta[t * TPAD + k] = v;
    }

    const unsigned int phi_off0 = (unsigned int)(uintptr_t)&phib[0][0];
    const unsigned int phi_off1 = (unsigned int)(uintptr_t)&phib[1][0];

    // Preload key tile 0 (fm part + rel-pos part = 2 descriptors).
    if (wave == 0) {
        tdm_load_tile(phi_off0,              fm_reference + 0, 256u);
        tdm_load_tile(phi_off0 + 256u * 64u, rpr + 0,          136u);
    }

    // Flash state: 8 rows per lane (row = 16*wave + i + 8*half).
    float m[8], d[8], ng[8], nb[8];
#pragma unroll
    for (int i = 0; i < 8; ++i) { m[i] = -INFINITY; d[i] = 0.f; ng[i] = 0.f; nb[i] = 0.f; }

    const float* ap = theta + (wave * 16 + l16) * TPAD + 2 * half;

    for (int j = 0; j < 256; ++j) {
        const int cur = j & 1;
        if (wave == 0) {
            if (j + 1 < 256) {
                const int sn = (j + 1) * 16;
                const unsigned int dst = cur ? phi_off0 : phi_off1;
                tdm_load_tile(dst,              fm_reference + sn, 256u);
                tdm_load_tile(dst + 256u * 64u, rpr + sn,          136u);
                __builtin_amdgcn_s_wait_tensorcnt(2);   // tile j resident
            } else {
                __builtin_amdgcn_s_wait_tensorcnt(0);
            }
        }
        __syncthreads();                                // tile j visible to all

        const float* bufp = &phib[cur][0];
        const int   s0   = j * 16;
        const int   scol = s0 + l16;
        const float pm   = 0.01f * mask_ref[b * HW + scol];
        const float gs   = gvec[b * HW + scol];
        const float bs   = bvec[b * HW + scol];

        v8f c = {0.f, 0.f, 0.f, 0.f, 0.f, 0.f, 0.f, 0.f};

        // Feature channels: phi scaled by 0.01*mask_ref[s] after LDS read.
        // unroll 2 -> compiler emits ds_load_2addr_b32 pairs for B (as in
        // the rel-pos loop) instead of scalar ds_load_b32 + address adds.
#pragma unroll 2
        for (int k0 = 0; k0 < 256; k0 += 4) {
            const v2f a = *(const v2f*)(ap + k0);
            const int k = k0 + 2 * half;
            v2f bb;
            bb.x = bufp[k * 16 + l16] * pm;
            bb.y = bufp[(k + 1) * 16 + l16] * pm;
            c = __builtin_amdgcn_wmma_f32_16x16x4_f32(
                    false, a, false, bb, (short)0, c, false, false);
        }
        // Rel-pos channels: raw.
#pragma unroll 2
        for (int k0 = 256; k0 < KTOT; k0 += 4) {
            const v2f a = *(const v2f*)(ap + k0);
            const int k = k0 + 2 * half;
            v2f bb;
            bb.x = bufp[k * 16 + l16];
            bb.y = bufp[(k + 1) * 16 + l16];
            c = __builtin_amdgcn_wmma_f32_16x16x4_f32(
                    false, a, false, bb, (short)0, c, false, false);
        }

        // Streaming softmax: denominator sums exp over ALL columns;
        // numerators drop columns where the raw score == 0 (reference's
        // torch.nonzero sparsification).
#pragma unroll
        for (int i = 0; i < 8; ++i) {
            const float sc = c[i];
            const float l  = 200.f * sc;
            const float tm = bf_max16(l);
            const float mn = fmaxf(m[i], tm);
            const float so = __expf(m[i] - mn);          // 0 on first tile
            const float e  = __expf(l - mn);
            const float dsum = bf_sum16(e);
            const float gsum = bf_sum16((sc != 0.f) ? e * gs : 0.f);
            const float bsum = bf_sum16((sc != 0.f) ? e * bs : 0.f);
            d[i]  = fmaf(d[i],  so, dsum);
            ng[i] = fmaf(ng[i], so, gsum);
            nb[i] = fmaf(nb[i], so, bsum);
            m[i]  = mn;
        }
        __syncthreads();          // everyone done reading buf before reuse
    }

    // Each wave owns its 16 rows completely -> direct contribution add.
    if (l16 == 0) {               // lanes 0 (rows i) and 16 (rows i+8)
#pragma unroll
        for (int i = 0; i < 8; ++i) {
            const int row = t0 + wave * 16 + i + 8 * half;
            atomicAdd(gamma + row, ng[i] / d[i]);
            atomicAdd(beta  + row, nb[i] / d[i]);
        }
    }
}

// ---------------------------------------------------------------------------
// Kernel 3: out = fm_source * (1 + gamma) + beta   (gamma/beta broadcast c)
// ---------------------------------------------------------------------------
__global__ __launch_bounds__(256) void amm_finish_kernel(
    const float* __restrict__ fm_source,
    const float* __restrict__ gamma,
    const float* __restrict__ beta,
    float* __restrict__ out)
{
    const int idx = blockIdx.x * 256 + threadIdx.x;   // 256*4096 elements
    const int s   = idx & (HW - 1);
    out[idx] = fmaf(fm_source[idx], 1.f + gamma[s], beta[s]);
}

// ---------------------------------------------------------------------------
extern "C" void kernel_launch(void* const* d_in, const int* in_sizes, int n_in,
                              void* d_out, int out_size, void* d_ws, size_t ws_size,
                              hipStream_t stream) {
    (void)in_sizes; (void)n_in; (void)out_size; (void)ws_size;
    const float* fm_source      = (const float*)d_in[0];
    const float* fm_reference   = (const float*)d_in[1];
    const float* mask_source    = (const float*)d_in[2];
    const float* mask_ref       = (const float*)d_in[3];
    const float* rel_pos_source = (const float*)d_in[4];
    const float* rel_pos_ref    = (const float*)d_in[5];
    const float* w_lambda       = (const float*)d_in[6];
    const float* w_beta         = (const float*)d_in[7];
    float* out = (float*)d_out;

    // Workspace (floats): gvec(3*4096) bvec(3*4096) gamma(4096) beta(4096)
    float* ws    = (float*)d_ws;
    float* gvec  = ws;
    float* bvec  = ws + 3 * HW;
    float* gamma = ws + 6 * HW;
    float* beta  = ws + 7 * HW;

    amm_prep_kernel<<<16, 256, 0, stream>>>(fm_reference, mask_ref, w_lambda, w_beta,
                                            gvec, bvec, gamma, beta);
    amm_attn_kernel<<<192, 128, 0, stream>>>(fm_source, fm_reference,
                                             mask_source, mask_ref,
                                             rel_pos_source, rel_pos_ref,
                                             gvec, bvec, gamma, beta);
    amm_finish_kernel<<<4096, 256, 0, stream>>>(fm_source, gamma, beta, out);
}